// Atcm_24438363914490
// MI455X (gfx1250) — compile-verified
//
#include <hip/hip_runtime.h>
#include <math.h>

// Problem constants (from reference): n=8, c=384, h=w=112, hid=24
#define NB   8
#define CC   384
#define HW   12544          // 112*112
#define HW4  3136           // HW / 4
#define HID  24
#define MSH  128            // m = n*dp shuffled channels
#define DP   16             // dp
#define GST  (CC + 4)       // gLds row stride (bank-conflict free: 388%64=4)
#define TST  36             // tLds row stride (36*M mod 64 distinct for M<16)

typedef float v2f __attribute__((ext_vector_type(2)));
typedef float v4f __attribute__((ext_vector_type(4)));
typedef float v8f __attribute__((ext_vector_type(8)));

// ---------------------------------------------------------------------------
// Pass 1: per-(sample, channel) sum and sum-of-squares of x.
// One block per (j,c) pair; 128-bit streaming loads; deterministic reduction.
// x stays resident in the 192MB L2 for the apply pass.
// ---------------------------------------------------------------------------
__global__ void __launch_bounds__(256)
stats_kernel(const float* __restrict__ x,
             float* __restrict__ sx, float* __restrict__ sq)
{
    const v4f* x4 = (const v4f*)x;
    size_t base = (size_t)blockIdx.x * HW4;
    float s = 0.f, q = 0.f;
    for (int t = threadIdx.x; t < HW4; t += 256) {
        v4f v = x4[base + t];
        s += v.x + v.y + v.z + v.w;
        q += v.x * v.x + v.y * v.y + v.z * v.z + v.w * v.w;
    }
    // wave32 butterfly reduction (deterministic)
    for (int m = 16; m >= 1; m >>= 1) {
        s += __shfl_xor(s, m, 32);
        q += __shfl_xor(q, m, 32);
    }
    __shared__ float ps[8], pq[8];
    int wave = threadIdx.x >> 5, lane = threadIdx.x & 31;
    if (lane == 0) { ps[wave] = s; pq[wave] = q; }
    __syncthreads();
    if (threadIdx.x == 0) {
        float S = 0.f, Q = 0.f;
        for (int i = 0; i < 8; ++i) { S += ps[i]; Q += pq[i]; }
        sx[blockIdx.x] = S;
        sq[blockIdx.x] = Q;
    }
}

// ---------------------------------------------------------------------------
// Pass 2 (single block, 256 threads = 8 waves):
//   phase 0: per-sample mu / rsqrt(var+eps) of y (pct-permuted channel sums)
//   phase 1: g[j][c] = mean_HW(z[j][c]) analytically, into LDS;
//            stage W1 into LDS (zero-padded to 32 cols) -> branch-free GEMM1
//   phase 2: t = relu(g @ W1)    via V_WMMA_F32_16X16X4_F32 (2 N-tiles)
//   phase 3: s = sigmoid(t @ W2) via WMMA (24 N-tiles over 8 waves)
// A-matrix 16x4 f32 layout: lanes 0-15 hold {K0,K1}, lanes 16-31 {K2,K3}.
// C/D layout: vgpr v, lanes 0-15 -> M=v, lanes 16-31 -> M=v+8; N=lane%16.
// ---------------------------------------------------------------------------
__global__ void __launch_bounds__(256)
se_kernel(const float* __restrict__ sx, const float* __restrict__ sq,
          const float* __restrict__ gamma, const float* __restrict__ beta,
          const float* __restrict__ w1, const float* __restrict__ w2,
          float* __restrict__ stats, float* __restrict__ svals)
{
    __shared__ float gLds[16][GST];   // A for GEMM1 (rows 8..15 zero pad)
    __shared__ float w1Lds[CC][32];   // B for GEMM1 (cols 24..31 zero)
    __shared__ float tLds[16][TST];   // A for GEMM2 (cols 24..31 are 0)
    __shared__ float muS[8], rsS[8];

    const int tid  = threadIdx.x;
    const int wave = tid >> 5;
    const int lane = tid & 31;
    const float invCHW = 1.f / ((float)CC * (float)HW);
    const float invHW  = 1.f / (float)HW;

    // ---- phase 0: per-sample stats of y (wave j handles sample j) ----
    if (wave < NB) {
        float sY = 0.f, sY2 = 0.f;
        for (int c2 = lane; c2 < CC; c2 += 32) {
            int si, scn;
            if (c2 < MSH) { si = c2 >> 4; scn = wave * DP + (c2 & 15); }
            else          { si = wave;    scn = c2; }
            sY  += sx[si * CC + scn];
            sY2 += sq[si * CC + scn];
        }
        for (int m = 16; m >= 1; m >>= 1) {
            sY  += __shfl_xor(sY,  m, 32);
            sY2 += __shfl_xor(sY2, m, 32);
        }
        if (lane == 0) {
            float mu  = sY * invCHW;
            float var = sY2 * invCHW - mu * mu;
            float rs  = rsqrtf(var + 1e-5f);
            muS[wave] = mu; rsS[wave] = rs;
            stats[wave]     = mu;
            stats[8 + wave] = rs;
        }
    }
    __syncthreads();

    // ---- phase 1a: g[j][c] = mean_HW(z) from channel sums ----
    for (int idx = tid; idx < 16 * CC; idx += 256) {
        int j = idx / CC, c = idx - j * CC;
        float gval = 0.f;
        if (j < NB) {
            int si, scn;
            if (c < MSH) { si = c >> 4; scn = j * DP + (c & 15); }
            else         { si = j;      scn = c; }
            gval = sx[j * CC + c] * invHW +
                   (sx[si * CC + scn] * invHW - muS[j]) * rsS[j] * gamma[c] +
                   beta[c];
        }
        gLds[j][c] = gval;
    }
    // ---- phase 1b: stage W1 into LDS, coalesced, zero-pad cols 24..31 ----
    for (int idx = tid; idx < CC * HID; idx += 256) {
        int r = idx / HID, cc = idx - r * HID;
        w1Lds[r][cc] = w1[idx];
    }
    for (int idx = tid; idx < CC * 8; idx += 256) {
        w1Lds[idx >> 3][HID + (idx & 7)] = 0.f;
    }
    __syncthreads();

    const int Mrow = lane & 15;
    const int koff = (lane < 16) ? 0 : 2;

    // ---- phase 2: t = relu(g @ W1) ----
    if (wave < 2) {
        v8f acc = {};
        const int nt  = wave;
        const int col = nt * 16 + (lane & 15);
        for (int k = 0; k < CC; k += 4) {
            v2f a, b;
            a.x = gLds[Mrow][k + koff];
            a.y = gLds[Mrow][k + koff + 1];
            b.x = w1Lds[k + koff][col];
            b.y = w1Lds[k + koff + 1][col];
            acc = __builtin_amdgcn_wmma_f32_16x16x4_f32(
                false, a, false, b, (short)0, acc, false, false);
        }
        for (int v = 0; v < 8; ++v) {
            int M = v + ((lane < 16) ? 0 : 8);
            tLds[M][nt * 16 + (lane & 15)] = fmaxf(acc[v], 0.f);
        }
    }
    __syncthreads();

    // ---- phase 3: s = sigmoid(t @ W2), W2 is [24][384] row-major ----
    for (int nt = wave; nt < CC / 16; nt += 8) {   // 24 tiles, 3 per wave
        v8f acc = {};
        const int col = nt * 16 + (lane & 15);
        for (int k = 0; k < HID; k += 4) {
            v2f a, b;
            a.x = tLds[Mrow][k + koff];
            a.y = tLds[Mrow][k + koff + 1];
            b.x = w2[(k + koff)     * CC + col];
            b.y = w2[(k + koff + 1) * CC + col];
            acc = __builtin_amdgcn_wmma_f32_16x16x4_f32(
                false, a, false, b, (short)0, acc, false, false);
        }
        if (lane < 16) {   // valid rows M=0..7 live entirely in lanes 0-15
            for (int v = 0; v < NB; ++v)
                svals[v * CC + col] = 1.f / (1.f + expf(-acc[v]));
        }
    }
}

// ---------------------------------------------------------------------------
// Pass 3: out = (x + (y - mu)*rs*gamma + beta) * s   fully fused, 128-bit IO.
// One block per (j,c). Shuffled source channel hits L2 (x resident after
// pass 1). Output stored non-temporally so it doesn't evict x from L2.
// ---------------------------------------------------------------------------
__global__ void __launch_bounds__(256)
apply_kernel(const float* __restrict__ x,
             const float* __restrict__ gamma, const float* __restrict__ beta,
             const float* __restrict__ stats, const float* __restrict__ svals,
             float* __restrict__ out)
{
    const int blk = blockIdx.x;
    const int j = blk / CC, c = blk - j * CC;
    const float mu = stats[j], rs = stats[8 + j];
    const float a  = rs * gamma[c];
    const float b  = beta[c] - mu * a;      // y*a + b == (y-mu)*rs*gamma + beta
    const float sc = svals[j * CC + c];

    int si, scn;
    if (c < MSH) { si = c >> 4; scn = j * DP + (c & 15); }
    else         { si = j;      scn = c; }

    const size_t base4 = (size_t)blk * HW4;
    const size_t src4  = ((size_t)si * CC + scn) * (size_t)HW4;

    const v4f* x4 = (const v4f*)x;
    v4f*       o4 = (v4f*)out;

    if (src4 == base4) {
        for (int t = threadIdx.x; t < HW4; t += 256) {
            v4f xv = x4[base4 + t];
            v4f r;
            r.x = (xv.x + fmaf(xv.x, a, b)) * sc;
            r.y = (xv.y + fmaf(xv.y, a, b)) * sc;
            r.z = (xv.z + fmaf(xv.z, a, b)) * sc;
            r.w = (xv.w + fmaf(xv.w, a, b)) * sc;
            __builtin_nontemporal_store(r, &o4[base4 + t]);
        }
    } else {
        for (int t = threadIdx.x; t < HW4; t += 256) {
            v4f xv = x4[base4 + t];
            v4f yv = x4[src4 + t];
            v4f r;
            r.x = (xv.x + fmaf(yv.x, a, b)) * sc;
            r.y = (xv.y + fmaf(yv.y, a, b)) * sc;
            r.z = (xv.z + fmaf(yv.z, a, b)) * sc;
            r.w = (xv.w + fmaf(yv.w, a, b)) * sc;
            __builtin_nontemporal_store(r, &o4[base4 + t]);
        }
    }
}

// ---------------------------------------------------------------------------
extern "C" void kernel_launch(void* const* d_in, const int* in_sizes, int n_in,
                              void* d_out, int out_size, void* d_ws, size_t ws_size,
                              hipStream_t stream)
{
    const float* x       = (const float*)d_in[0];
    const float* gweight = (const float*)d_in[1];
    const float* gbias   = (const float*)d_in[2];
    const float* w1      = (const float*)d_in[3];
    const float* w2      = (const float*)d_in[4];
    float*       out     = (float*)d_out;

    float* ws    = (float*)d_ws;
    float* sx    = ws;                 // 3072: per-(n,c) sum of x
    float* sq    = ws + NB * CC;       // 3072: per-(n,c) sumsq of x
    float* stats = ws + 2 * NB * CC;   // 16:   mu[8], rs[8]
    float* svals = stats + 16;         // 3072: SE scale s[n,c]

    stats_kernel<<<NB * CC, 256, 0, stream>>>(x, sx, sq);
    se_kernel<<<1, 256, 0, stream>>>(sx, sq, gweight, gbias, w1, w2, stats, svals);
    apply_kernel<<<NB * CC, 256, 0, stream>>>(x, gweight, gbias, stats, svals, out);
}